// Apply_on_single_area_59897613910381
// MI455X (gfx1250) — compile-verified
//
#include <hip/hip_runtime.h>
#include <hip/hip_bf16.h>
#include <stdint.h>

// ---------------- configuration ----------------
#define DIAM 32
#define NTHREADS 64                   // 2 waves (wave32) per block
#define WPB (NTHREADS / 32)

#define STG_STRIDE_F 68               // staged raw mask row stride (272B: 16B-aligned, low conflict)
#define STG_WAVE_F   (DIAM * STG_STRIDE_F)   // 2176 floats
#define MB_STRIDE_F  33               // filtered-mask row stride (conflict-free column access)
#define MB_WAVE_F    (DIAM * MB_STRIDE_F)    // 1056 floats
#define IMG_STRIDE_F 36               // image row stride (144B: 16B-aligned)
#define IMG_WAVE_F   (DIAM * IMG_STRIDE_F)   // 1152 floats
#define LDS_WAVE_F   (STG_WAVE_F + MB_WAVE_F + IMG_WAVE_F)  // 4384 floats = 17536 B / wave

#define INV_2PI 0.15915494309189535f
#define EPSF    1e-8f

typedef int v4i __attribute__((vector_size(16)));

// ---- differentiable rounding primitives ----
// v_sin_f32 computes sin(2*pi*x) natively -> diff_round(x) = x - sin(2pi x)/(2pi)
static __device__ __forceinline__ float dround(float x) {
  return fmaf(-__builtin_amdgcn_sinf(x), INV_2PI, x);
}
static __device__ __forceinline__ float hdround(float x) { return dround(dround(x)); }

// differentiable_and(a, b) with hb = hdround(b) pre-hoisted (b uniform per area)
static __device__ __forceinline__ float dand_pre(float a, float hb) {
  float ha = hdround(a);
  return hdround(fmaf(ha, hb, (1.0f - ha) * (1.0f - hb)));
}
static __device__ __forceinline__ float dand_full(float a, float b) {
  float ha = hdround(a);
  float hb = hdround(b);
  return hdround(fmaf(ha, hb, (1.0f - ha) * (1.0f - hb)));
}

// ---- wave32 butterfly reduction ----
static __device__ __forceinline__ float wave_sum(float v) {
#pragma unroll
  for (int off = 16; off > 0; off >>= 1) v += __shfl_xor(v, off, 32);
  return v;
}

// ---- CDNA5 sync helpers ----
static __device__ __forceinline__ void lds_fence_wave() {
  asm volatile("s_wait_dscnt 0x0" ::: "memory");
}

static __device__ __forceinline__ void wait_async0() {
#if __has_builtin(__builtin_amdgcn_s_wait_asynccnt)
  __builtin_amdgcn_s_wait_asynccnt(0);
  asm volatile("" ::: "memory");
#else
  asm volatile("s_wait_asynccnt 0x0" ::: "memory");
#endif
}

// ---- async global -> LDS 16B/lane DMA (ASYNCcnt path) ----
static __device__ __forceinline__ void async_b128(const void* g, void* l) {
#if __has_builtin(__builtin_amdgcn_global_load_async_to_lds_b128)
  // clang signature: (v4i32* global_src, v4i32* lds_dst, imm offset, imm cpol)
  __builtin_amdgcn_global_load_async_to_lds_b128((v4i*)g, (v4i*)l, 0, 0);
#else
  unsigned int loff = (unsigned int)(uintptr_t)l;
  unsigned long long ga = (unsigned long long)(uintptr_t)g;
  asm volatile("global_load_async_to_lds_b128 %0, %1, off" :: "v"(loff), "v"(ga) : "memory");
#endif
}

// Stage one area's mask tile (32x32x2 f32 = 8 KB, rows of 256B) into padded LDS.
// Each instruction moves 512B (2 rows); LDS rows padded to 272B, so bump pointers per step.
static __device__ __forceinline__ void stage_mask_async(const float* gmask, float* stage, int lane) {
  const char* g = (const char*)gmask + ((lane >> 4) * 256 + (lane & 15) * 16);
  char* l = (char*)stage + ((lane >> 4) * (STG_STRIDE_F * 4) + (lane & 15) * 16);
#pragma unroll
  for (int k = 0; k < 16; ++k) {
    async_b128(g, l);
    g += 512;
    l += 2 * STG_STRIDE_F * 4;   // 544
  }
}

// Stage one area's image tile (32x32 f32 = 4 KB, rows of 128B) into padded LDS.
static __device__ __forceinline__ void stage_img_async(const float* gimg, float* imgbuf, int lane) {
  const char* g = (const char*)gimg + ((lane >> 3) * 128 + (lane & 7) * 16);
  char* l = (char*)imgbuf + ((lane >> 3) * (IMG_STRIDE_F * 4) + (lane & 7) * 16);
#pragma unroll
  for (int k = 0; k < 8; ++k) {
    async_b128(g, l);
    g += 512;
    l += 4 * IMG_STRIDE_F * 4;   // 576
  }
}

// Phase 1: filtered mask m = dand(dand(m0,sx), dand(m1,sy)) for this lane's row.
// Partial unroll: 4 independent sin-chains give enough ILP for the quarter-rate
// trans unit without blowing up VGPR pressure (full unroll pushed us past v256).
static __device__ __forceinline__ void compute_m_rows(const float* stage, float* mbuf,
                                                      float hx, float hy, int lane) {
  const float2* srow = (const float2*)(stage + lane * STG_STRIDE_F);
  float* mrow = mbuf + lane * MB_STRIDE_F;
#pragma unroll 4
  for (int j = 0; j < DIAM; ++j) {
    float2 ab = srow[j];
    float c0 = dand_pre(ab.x, hx);
    float c1 = dand_pre(ab.y, hy);
    mrow[j] = dand_full(c0, c1);
  }
}

// mean/var of image under a translated mask; `shifted(j)` supplies the shifted mask value.
template <typename F>
static __device__ __forceinline__ void feat_pair(const float* mrow, const float* irow,
                                                 F shifted, float* mean_out, float* var_out) {
  float s0 = 0.f, s1 = 0.f;
#pragma unroll 8
  for (int j = 0; j < DIAM; ++j) {
    float tm = mrow[j] * shifted(j);
    s0 += tm;
    s1 += irow[j] * tm;
  }
  float msum = wave_sum(s0) + EPSF;
  float mean = wave_sum(s1) / msum;
  float s2 = 0.f;
#pragma unroll 8
  for (int j = 0; j < DIAM; ++j) {
    float tm = mrow[j] * shifted(j);
    float d  = fmaf(irow[j], tm, -mean) * tm;
    s2 = fmaf(d, d, s2);
  }
  *mean_out = mean;
  *var_out  = wave_sum(s2) / msum;
}

static __device__ __forceinline__ float translated_mask_variance_w(const float* mbuf,
                                                                   const float* imgbuf, int lane) {
  const float* mrow = mbuf + lane * MB_STRIDE_F;
  const float* irow = imgbuf + lane * IMG_STRIDE_F;
  float means[4], vars[4];

  { // axis 0, backward: shifted[i] = m[i+2]
    int rs = lane + 2;
    float vm = (rs < DIAM) ? 1.f : 0.f;
    const float* srow = mbuf + (rs < DIAM ? rs : DIAM - 1) * MB_STRIDE_F;
    feat_pair(mrow, irow, [&](int j) { return srow[j] * vm; }, &means[0], &vars[0]);
  }
  { // axis 0, forward: shifted[i] = m[i-2]
    int rs = lane - 2;
    float vm = (rs >= 0) ? 1.f : 0.f;
    const float* srow = mbuf + (rs >= 0 ? rs : 0) * MB_STRIDE_F;
    feat_pair(mrow, irow, [&](int j) { return srow[j] * vm; }, &means[1], &vars[1]);
  }
  { // axis 1, backward: shifted[.,j] = m[.,j+2]
    feat_pair(mrow, irow, [&](int j) { int js = j + 2; return (js < DIAM) ? mrow[js] : 0.f; },
              &means[2], &vars[2]);
  }
  { // axis 1, forward: shifted[.,j] = m[.,j-2]
    feat_pair(mrow, irow, [&](int j) { int js = j - 2; return (js >= 0) ? mrow[js] : 0.f; },
              &means[3], &vars[3]);
  }

  float fm = 0.25f * (means[0] + means[1] + means[2] + means[3]);
  float a0 = means[0] - fm, a1 = means[1] - fm, a2 = means[2] - fm, a3 = means[3] - fm;
  float vmean = 0.25f * (a0 * a0 + a1 * a1 + a2 * a2 + a3 * a3);
  float gv = 0.25f * (vars[0] + vars[1] + vars[2] + vars[3]);
  float b0 = vars[0] - gv, b1 = vars[1] - gv, b2 = vars[2] - gv, b3 = vars[3] - gv;
  float vvar = 0.25f * (b0 * b0 + b1 * b1 + b2 * b2 + b3 * b3);
  return 0.5f * (vmean + vvar);
}

__global__ __launch_bounds__(NTHREADS)
__attribute__((amdgpu_waves_per_eu(5)))
void per_area_loss_kernel(const float* __restrict__ image, const float* __restrict__ mask,
                          const float* __restrict__ mask_alt, const int* __restrict__ mask_index,
                          float* __restrict__ out, int n_areas) {
  extern __shared__ float lds[];
  const int lane = threadIdx.x & 31;
  const int wv   = threadIdx.x >> 5;
  const int area = blockIdx.x * WPB + wv;
  if (area >= n_areas) return;   // wave-uniform; no block barriers anywhere

  float* stage  = lds + wv * LDS_WAVE_F;
  float* mbuf   = stage + STG_WAVE_F;
  float* imgbuf = mbuf + MB_WAVE_F;

  const size_t mask_off = (size_t)area * (DIAM * DIAM * 2);
  const size_t img_off  = (size_t)area * (DIAM * DIAM);

  // Kick off async DMA for main mask + image tiles (ASYNCcnt).
  stage_mask_async(mask + mask_off, stage, lane);
  stage_img_async(image + img_off, imgbuf, lane);

  // Uniform per-area shift bits (overlapped with DMA).
  int idx = mask_index[area];
  float hx = hdround((float)(idx & 1));
  float hy = hdround((float)(idx >> 1));

  wait_async0();
  compute_m_rows(stage, mbuf, hx, hy, lane);        // sin-heavy
  lds_fence_wave();                                  // staging reads + m stores drained

  // Prefetch alt mask into freed staging buffer, overlapped with main feature pass.
  stage_mask_async(mask_alt + mask_off, stage, lane);
  float loss_main = translated_mask_variance_w(mbuf, imgbuf, lane);

  wait_async0();
  compute_m_rows(stage, mbuf, hx, hy, lane);        // alt mask
  lds_fence_wave();
  float loss_alt = translated_mask_variance_w(mbuf, imgbuf, lane);

  if (lane == 0) out[area] = loss_main / (loss_main + loss_alt + EPSF);
}

extern "C" void kernel_launch(void* const* d_in, const int* in_sizes, int n_in,
                              void* d_out, int out_size, void* d_ws, size_t ws_size,
                              hipStream_t stream) {
  const float* image    = (const float*)d_in[0];
  const float* mask     = (const float*)d_in[1];
  const float* mask_alt = (const float*)d_in[2];
  const int*   midx     = (const int*)d_in[3];
  float* out = (float*)d_out;
  (void)d_ws; (void)ws_size; (void)n_in; (void)out_size;

  int n = in_sizes[3];
  int blocks = (n + WPB - 1) / WPB;
  size_t shmem = (size_t)WPB * LDS_WAVE_F * sizeof(float);   // 35072 B
  per_area_loss_kernel<<<blocks, NTHREADS, shmem, stream>>>(image, mask, mask_alt, midx, out, n);
}